// Q0DistMultiTask_81922206204287
// MI455X (gfx1250) — compile-verified
//
#include <hip/hip_runtime.h>
#include <math.h>

// ---------------------------------------------------------------------------
// Problem constants (from the reference)
// ---------------------------------------------------------------------------
#define T_WIN   48                 // time points
#define N_TASK  128                // tasks
#define DTOT    (T_WIN * N_TASK)   // 6144
#define NB      128                // Cholesky block size
#define NBLK    (DTOT / NB)        // 48 blocks
#define RANK_   8
#define NSAMP   64
#define PI_F    3.14159265358979323846f
#define SP      20                 // LDS row stride (floats): 16B-aligned rows,
                                   // conflict-free (banks 4*(5r mod 16) +/- kk)

typedef __attribute__((ext_vector_type(2))) float v2f;
typedef __attribute__((ext_vector_type(8))) float v8f;

__device__ __forceinline__ v8f v8f_zero() {
    v8f z = {0.f, 0.f, 0.f, 0.f, 0.f, 0.f, 0.f, 0.f};
    return z;
}

// fp32 WMMA: D(16x16,f32) = A(16x4,f32) x B(4x16,f32) + C
__device__ __forceinline__ v8f wmma_f32_16x16x4(v2f a, v2f b, v8f c) {
    return __builtin_amdgcn_wmma_f32_16x16x4_f32(false, a, false, b,
                                                 (short)0, c, false, false);
}

// CDNA5 async copy: global -> LDS, 16B per lane, tracked by ASYNCcnt.
// addrspace(3) pointers are 32-bit LDS offsets; generic pointers to LDS carry
// the offset in their low 32 bits, which is what the VDST VGPR wants.
__device__ __forceinline__ void async_load_b128(void* lds_ptr, const float* gaddr) {
    unsigned int lds_off = (unsigned int)(unsigned long long)lds_ptr;
    asm volatile("global_load_async_to_lds_b128 %0, %1, off"
                 :: "v"(lds_off), "v"(gaddr)
                 : "memory");
}

__device__ __forceinline__ void wait_async0() {
#if defined(__has_builtin) && __has_builtin(__builtin_amdgcn_s_wait_asynccnt)
    __builtin_amdgcn_s_wait_asynccnt(0);
#else
    asm volatile("s_wait_asynccnt 0x0" ::: "memory");
#endif
}

// ---------------------------------------------------------------------------
// Kernel 1: K_t (48x48 RBF) and B = F F^T + diag(softplus(raw_var))
// ---------------------------------------------------------------------------
__global__ void k_build_small(const float* __restrict__ cf,
                              const float* __restrict__ raw_var,
                              const float* __restrict__ raw_ls,
                              float* __restrict__ Kt,
                              float* __restrict__ Bm) {
    float ls = log1pf(expf(raw_ls[0]));  // softplus
    for (int idx = threadIdx.x; idx < T_WIN * T_WIN; idx += blockDim.x) {
        int i = idx / T_WIN, j = idx % T_WIN;
        float d = (float)(i - j) * (PI_F / 24.0f) / ls;
        Kt[idx] = expf(-0.5f * d * d);
    }
    for (int idx = threadIdx.x; idx < N_TASK * N_TASK; idx += blockDim.x) {
        int i = idx / N_TASK, j = idx % N_TASK;
        float s = 0.f;
        for (int r = 0; r < RANK_; ++r) s += cf[i * RANK_ + r] * cf[j * RANK_ + r];
        if (i == j) s += log1pf(expf(raw_var[i]));
        Bm[idx] = s;
    }
}

// ---------------------------------------------------------------------------
// Kernel 2: Sigma = kron(K_t, B) + diag(task_noise + noise + jitter)
// ---------------------------------------------------------------------------
__global__ void k_build_sigma(const float* __restrict__ Kt,
                              const float* __restrict__ Bm,
                              const float* __restrict__ task_noises,
                              const float* __restrict__ noise,
                              float* __restrict__ S) {
    long long gid = (long long)blockIdx.x * blockDim.x + threadIdx.x;
    if (gid >= (long long)DTOT * DTOT) return;
    int r = (int)(gid / DTOT), c = (int)(gid % DTOT);
    int ti = r / N_TASK, a = r % N_TASK;
    int tj = c / N_TASK, b = c % N_TASK;
    float v = Kt[ti * T_WIN + tj] * Bm[a * N_TASK + b];
    if (r == c) v += task_noises[a] + noise[0] + 1e-6f;
    S[gid] = v;
}

// ---------------------------------------------------------------------------
// Kernel 3: factor 128x128 diagonal block in LDS + build Linv = L_kk^{-1}
// (single workgroup, 128 threads = 4 wave32s)
// ---------------------------------------------------------------------------
__global__ void k_potrf_diag(float* __restrict__ S, float* __restrict__ Linv, int k) {
    __shared__ float A[NB * NB];  // 64 KB of the 320 KB WGP LDS
    int t = threadIdx.x;
    float* G = S + (long long)(k * NB) * DTOT + (long long)k * NB;
    for (int j = 0; j < NB; ++j) A[t * NB + j] = G[(long long)t * DTOT + j];
    __syncthreads();
    for (int c = 0; c < NB; ++c) {
        float dg = sqrtf(A[c * NB + c]);   // every thread reads pre-update diag
        __syncthreads();
        if (t == c)      A[c * NB + c] = dg;
        else if (t > c)  A[t * NB + c] /= dg;
        __syncthreads();
        if (t > c) {                        // rank-1 update of lower triangle
            float lic = A[t * NB + c];
            for (int j = c + 1; j <= t; ++j) A[t * NB + j] -= lic * A[j * NB + c];
        }
        __syncthreads();
    }
    // write back L_kk (lower incl. diagonal)
    for (int j = 0; j <= t; ++j) G[(long long)t * DTOT + j] = A[t * NB + j];
    // forward substitution: thread t solves column t of L_kk * X = I
    for (int r = 0; r < NB; ++r) {
        float x;
        if (r < t) {
            x = 0.f;
        } else {
            float s = (r == t) ? 1.f : 0.f;
            for (int j = t; j < r; ++j) s -= A[r * NB + j] * Linv[j * NB + t];
            x = s / A[r * NB + r];
        }
        Linv[r * NB + t] = x;  // thread-private column: RAW within same thread only
    }
}

// ---------------------------------------------------------------------------
// Kernel 4: panel solve as GEMM:  X_i = A_ik * Linv^T   (one WG per block row)
// 256 threads = 8 waves; wave w owns a 16x128 strip (8 accumulator tiles).
// Double-buffered async (global->LDS) staging overlapped with WMMA.
// ---------------------------------------------------------------------------
__global__ void k_panel(float* __restrict__ S, const float* __restrict__ Linv, int k) {
    __shared__ __attribute__((aligned(16))) float sA[2][NB * SP];
    __shared__ __attribute__((aligned(16))) float sL[2][NB * SP];
    int i = k + 1 + blockIdx.x;
    int w = threadIdx.x >> 5, lane = threadIdx.x & 31;
    int lrow = lane & 15, lhi = lane >> 4;
    const float* Ablk = S + (long long)(i * NB) * DTOT + (long long)k * NB;

    v8f acc[8];
    for (int c = 0; c < 8; ++c) acc[c] = v8f_zero();

    // issue chunk 0
    for (int t = 0; t < 2; ++t) {
        int qd = threadIdx.x + t * 256;       // 0..511 quads (128 rows x 4)
        int rr = qd >> 2, qq = qd & 3;
        async_load_b128(&sA[0][rr * SP + qq * 4],
                        Ablk + (long long)rr * DTOT + qq * 4);
        async_load_b128(&sL[0][rr * SP + qq * 4],
                        Linv + rr * NB + qq * 4);
    }
    for (int kb = 0; kb < NB / 16; ++kb) {
        int cur = kb & 1;
        wait_async0();
        __syncthreads();
        if (kb + 1 < NB / 16) {               // prefetch next chunk
            int nb = cur ^ 1, kn = (kb + 1) * 16;
            for (int t = 0; t < 2; ++t) {
                int qd = threadIdx.x + t * 256;
                int rr = qd >> 2, qq = qd & 3;
                async_load_b128(&sA[nb][rr * SP + qq * 4],
                                Ablk + (long long)rr * DTOT + kn + qq * 4);
                async_load_b128(&sL[nb][rr * SP + qq * 4],
                                Linv + rr * NB + kn + qq * 4);
            }
        }
        for (int kk = 0; kk < 16; kk += 4) {
            int kq = kk + 2 * lhi;
            v2f a;
            a.x = sA[cur][(w * 16 + lrow) * SP + kq];
            a.y = sA[cur][(w * 16 + lrow) * SP + kq + 1];
            for (int c = 0; c < 8; ++c) {
                v2f b;                         // B[k][n] = Linv[n][k]
                b.x = sL[cur][(c * 16 + lrow) * SP + kq];
                b.y = sL[cur][(c * 16 + lrow) * SP + kq + 1];
                acc[c] = wmma_f32_16x16x4(a, b, acc[c]);
            }
        }
    }
    // overwrite panel in place (all reads of Ablk done before last barrier)
    float* Out = S + (long long)(i * NB) * DTOT + (long long)k * NB;
    for (int c = 0; c < 8; ++c)
        for (int e = 0; e < 8; ++e) {
            int m = w * 16 + e + 8 * lhi;      // C layout: VGPR e -> M=e (+8 hi half)
            int n = c * 16 + lrow;
            Out[(long long)m * DTOT + n] = acc[c][e];
        }
}

// ---------------------------------------------------------------------------
// Kernel 5: trailing SYRK/GEMM update: A_ij -= L_ik * L_jk^T  (i >= j > k)
// One WG per block pair; same double-buffered async staging as k_panel.
// ---------------------------------------------------------------------------
__global__ void k_syrk(float* __restrict__ S, int k) {
    __shared__ __attribute__((aligned(16))) float sA[2][NB * SP];
    __shared__ __attribute__((aligned(16))) float sB[2][NB * SP];
    int p = blockIdx.x;
    int ii = (int)((sqrtf(8.f * (float)p + 1.f) - 1.f) * 0.5f);
    while ((ii + 1) * (ii + 2) / 2 <= p) ++ii;
    while (ii * (ii + 1) / 2 > p) --ii;
    int jj = p - ii * (ii + 1) / 2;
    int i = k + 1 + ii, j = k + 1 + jj;

    int w = threadIdx.x >> 5, lane = threadIdx.x & 31;
    int lrow = lane & 15, lhi = lane >> 4;
    const float* Ai = S + (long long)(i * NB) * DTOT + (long long)k * NB;
    const float* Aj = S + (long long)(j * NB) * DTOT + (long long)k * NB;

    v8f acc[8];
    for (int c = 0; c < 8; ++c) acc[c] = v8f_zero();

    for (int t = 0; t < 2; ++t) {
        int qd = threadIdx.x + t * 256;
        int rr = qd >> 2, qq = qd & 3;
        async_load_b128(&sA[0][rr * SP + qq * 4], Ai + (long long)rr * DTOT + qq * 4);
        async_load_b128(&sB[0][rr * SP + qq * 4], Aj + (long long)rr * DTOT + qq * 4);
    }
    for (int kb = 0; kb < NB / 16; ++kb) {
        int cur = kb & 1;
        wait_async0();
        __syncthreads();
        if (kb + 1 < NB / 16) {
            int nb = cur ^ 1, kn = (kb + 1) * 16;
            for (int t = 0; t < 2; ++t) {
                int qd = threadIdx.x + t * 256;
                int rr = qd >> 2, qq = qd & 3;
                async_load_b128(&sA[nb][rr * SP + qq * 4],
                                Ai + (long long)rr * DTOT + kn + qq * 4);
                async_load_b128(&sB[nb][rr * SP + qq * 4],
                                Aj + (long long)rr * DTOT + kn + qq * 4);
            }
        }
        for (int kk = 0; kk < 16; kk += 4) {
            int kq = kk + 2 * lhi;
            v2f a;
            a.x = sA[cur][(w * 16 + lrow) * SP + kq];
            a.y = sA[cur][(w * 16 + lrow) * SP + kq + 1];
            for (int c = 0; c < 8; ++c) {
                v2f b;                         // B[k][n] = Aj[n][k]
                b.x = sB[cur][(c * 16 + lrow) * SP + kq];
                b.y = sB[cur][(c * 16 + lrow) * SP + kq + 1];
                acc[c] = wmma_f32_16x16x4(a, b, acc[c]);
            }
        }
    }
    float* C = S + (long long)(i * NB) * DTOT + (long long)j * NB;
    for (int c = 0; c < 8; ++c)
        for (int e = 0; e < 8; ++e) {
            int m = w * 16 + e + 8 * lhi;
            int n = c * 16 + lrow;
            C[(long long)m * DTOT + n] -= acc[c][e];
        }
}

// ---------------------------------------------------------------------------
// Kernel 6: samples = eps @ L^T (L = lower triangle of S; mask j > i because
// the upper triangle still holds stale Sigma values).
// One WG per 128-wide output block column; 8 waves; each wave does 4 tiles.
// ---------------------------------------------------------------------------
__global__ void k_sample(const float* __restrict__ S,
                         const float* __restrict__ eps,
                         float* __restrict__ out) {
    int bi = blockIdx.x;
    int w = threadIdx.x >> 5, lane = threadIdx.x & 31;
    int lrow = lane & 15, lhi = lane >> 4;
    int sr = w & 3;   // sample tile row (samples sr*16 .. sr*16+15)
    int h  = w >> 2;  // which half of the 8 column tiles

    v8f acc[4];
    for (int c = 0; c < 4; ++c) acc[c] = v8f_zero();

    int kend = (bi + 1) * NB;  // columns j <= block-row end contribute
    for (int j0 = 0; j0 < kend; j0 += 4) {
        int kq = j0 + 2 * lhi;
        v2f a;
        a.x = eps[(sr * 16 + lrow) * DTOT + kq];
        a.y = eps[(sr * 16 + lrow) * DTOT + kq + 1];
        for (int c = 0; c < 4; ++c) {
            int icol = bi * NB + (h * 4 + c) * 16 + lrow;  // row index of L
            v2f b;                                          // B[k][n] = L[icol][j]
            b.x = (kq     <= icol) ? S[(long long)icol * DTOT + kq]     : 0.f;
            b.y = (kq + 1 <= icol) ? S[(long long)icol * DTOT + kq + 1] : 0.f;
            acc[c] = wmma_f32_16x16x4(a, b, acc[c]);
        }
    }
    for (int c = 0; c < 4; ++c)
        for (int e = 0; e < 8; ++e) {
            int s = sr * 16 + e + 8 * lhi;
            int n = bi * NB + (h * 4 + c) * 16 + lrow;
            out[(long long)s * DTOT + n] = acc[c][e];
        }
}

// ---------------------------------------------------------------------------
// Host launch: build Sigma in d_ws (151 MB; fits the 192 MB L2), 48-step
// blocked Cholesky, then the triangular sampling GEMM. All on `stream`.
// ---------------------------------------------------------------------------
extern "C" void kernel_launch(void* const* d_in, const int* in_sizes, int n_in,
                              void* d_out, int out_size, void* d_ws, size_t ws_size,
                              hipStream_t stream) {
    (void)in_sizes; (void)n_in; (void)out_size; (void)ws_size;
    const float* eps         = (const float*)d_in[0];  // [64, 6144]
    const float* covar       = (const float*)d_in[1];  // [128, 8]
    const float* raw_var     = (const float*)d_in[2];  // [128]
    const float* raw_ls      = (const float*)d_in[3];  // [1]
    const float* task_noises = (const float*)d_in[4];  // [128]
    const float* noise       = (const float*)d_in[5];  // [1]
    float* out = (float*)d_out;                        // [64, 6144]

    // workspace layout (floats): Sigma | Linv | Kt | B
    float* Sg   = (float*)d_ws;
    size_t off  = (size_t)DTOT * DTOT;
    float* Linv = Sg + off; off += (size_t)NB * NB;
    float* Kt   = Sg + off; off += (size_t)T_WIN * T_WIN;
    float* Bm   = Sg + off;

    k_build_small<<<1, 256, 0, stream>>>(covar, raw_var, raw_ls, Kt, Bm);

    long long tot = (long long)DTOT * DTOT;
    int ngrid = (int)((tot + 255) / 256);
    k_build_sigma<<<ngrid, 256, 0, stream>>>(Kt, Bm, task_noises, noise, Sg);

    for (int k = 0; k < NBLK; ++k) {
        k_potrf_diag<<<1, NB, 0, stream>>>(Sg, Linv, k);
        int r = NBLK - 1 - k;
        if (r > 0) {
            k_panel<<<r, 256, 0, stream>>>(Sg, Linv, k);
            k_syrk<<<r * (r + 1) / 2, 256, 0, stream>>>(Sg, k);
        }
    }
    k_sample<<<NBLK, 256, 0, stream>>>(Sg, eps, out);
}